// FrFT2DHSI_91001767068364
// MI455X (gfx1250) — compile-verified
//
#include <hip/hip_runtime.h>
#include <math.h>

#define NN 256
#define NB 4
#define NC 32
#define NSLICE (NB * NC)          /* 128 slices of 256x256 */
#define SLICE_ELEMS (NN * NN)     /* 65536 */
#define PI_F 3.14159265358979323846f

typedef __attribute__((ext_vector_type(2))) float v2f;
typedef __attribute__((ext_vector_type(8))) float v8f;

// D = A(16x4) * B(4x16) + C, full f32. f32 WMMA has no A/B NEG bits, so complex
// subtraction is done by negating the A fragment in VALU before the call.
__device__ __forceinline__ v8f wmma_f32(v2f a, v2f b, v8f c) {
  return __builtin_amdgcn_wmma_f32_16x16x4_f32(
      /*neg_a=*/false, a, /*neg_b=*/false, b,
      /*c_mod=*/(short)0, c, /*reuse_a=*/false, /*reuse_b=*/false);
}

// ---------------------------------------------------------------------------
// Kernel 1: build the 256x256 complex FRFT operator M (closed form).
// M[p,q] = c1[p~]*c1[q~] * (1/N) * sum_k c2[k] e^{2pi i k (p~-q~)/N} / sqrt(|sin a|+eps)
// ---------------------------------------------------------------------------
__global__ __launch_bounds__(256) void build_M_kernel(
    const float* __restrict__ alpha_p,
    float* __restrict__ Mr, float* __restrict__ Mi) {
  int idx = blockIdx.x * blockDim.x + threadIdx.x;
  if (idx >= SLICE_ELEMS) return;
  int p = idx >> 8;
  int q = idx & 255;

  float a = alpha_p[0];
  a = fminf(fmaxf(a, 1e-4f), 2.0f - 1e-4f);
  float s = sinf(a * (PI_F * 0.5f));
  float t = tanf(a * (PI_F * 0.25f));

  int pt = (p + 128) & 255;
  int qt = (q + 128) & 255;
  int d = pt - qt;
  float np_ = (float)(pt - 128);
  float nq_ = (float)(qt - 128);

  // Chirp-DFT sum: sum_k exp(i*(-pi*(k-128)^2/(N s) + 2 pi k d / N))
  float inv_ns = 1.0f / ((float)NN * s);
  float w0 = (2.0f * PI_F / (float)NN) * (float)d;
  float accr = 0.0f, acci = 0.0f;
  for (int k = 0; k < NN; ++k) {
    float km = (float)(k - 128);
    float ph = -PI_F * km * km * inv_ns + w0 * (float)k;
    float sp, cp;
    __sincosf(ph, &sp, &cp);
    accr += cp;
    acci += sp;
  }

  // outer chirps c1[p~]*c1[q~] = exp(-i pi t (np^2+nq^2)/N)
  float phc = -PI_F * t * (np_ * np_ + nq_ * nq_) * (1.0f / (float)NN);
  float sc, cc;
  __sincosf(phc, &sc, &cc);

  float scale = 1.0f / ((float)NN * sqrtf(fabsf(s) + 1e-12f));
  Mr[idx] = (cc * accr - sc * acci) * scale;
  Mi[idx] = (cc * acci + sc * accr) * scale;
}

// ---------------------------------------------------------------------------
// Pass 1: Y1 = X * M^T per slice (X real).  Y1[h,w] = sum_k X[h,k] * M[w,k].
// One wave32 computes a 16x16 tile of both Y1r and Y1i (2 WMMAs per k-step).
// ---------------------------------------------------------------------------
__global__ __launch_bounds__(128) void frft_pass1(
    const float* __restrict__ X,
    const float* __restrict__ Mr, const float* __restrict__ Mi,
    float* __restrict__ Y1r, float* __restrict__ Y1i) {
  const int lane = threadIdx.x & 31;
  const int wave = threadIdx.x >> 5;
  const int tile = blockIdx.x * 4 + wave;  // 256 tiles per slice
  const int slice = blockIdx.y;
  const int th = (tile >> 4) << 4;  // output row block (h)
  const int tw = (tile & 15) << 4;  // output col block (w)
  const int m = lane & 15;          // row within fragment / col for B,C,D
  const int kg = lane >> 4;         // k-pair group

  const float* Xs = X + (size_t)slice * SLICE_ELEMS;
  v8f cr = {};
  v8f ci = {};
  for (int k = 0; k < NN; k += 4) {
    const int ko = k + 2 * kg;
    // A: X[th+m][ko..ko+1] (16x4 fragment)
    v2f a = *(const v2f*)(Xs + (th + m) * NN + ko);
    // B: B[kk][n] = M[tw+n][k+kk]; rows of M are contiguous -> v2f loads
    v2f br = *(const v2f*)(Mr + (tw + m) * NN + ko);
    v2f bi = *(const v2f*)(Mi + (tw + m) * NN + ko);
    cr = wmma_f32(a, br, cr);
    ci = wmma_f32(a, bi, ci);
  }

  float* y1r = Y1r + (size_t)slice * SLICE_ELEMS;
  float* y1i = Y1i + (size_t)slice * SLICE_ELEMS;
#pragma unroll
  for (int r = 0; r < 8; ++r) {
    const int row = th + r + 8 * kg;
    y1r[row * NN + tw + m] = cr[r];
    y1i[row * NN + tw + m] = ci[r];
  }
}

// ---------------------------------------------------------------------------
// Pass 2: Y2 = M * Y1 per slice (complex x complex, 4 WMMAs per k-step).
// Y2r = Mr*Br + (-Mi)*Bi ; Y2i = Mr*Bi + Mi*Br
// ---------------------------------------------------------------------------
__global__ __launch_bounds__(128) void frft_pass2(
    const float* __restrict__ Mr, const float* __restrict__ Mi,
    const float* __restrict__ Y1r, const float* __restrict__ Y1i,
    float* __restrict__ Y2r, float* __restrict__ Y2i) {
  const int lane = threadIdx.x & 31;
  const int wave = threadIdx.x >> 5;
  const int tile = blockIdx.x * 4 + wave;
  const int slice = blockIdx.y;
  const int th = (tile >> 4) << 4;
  const int tw = (tile & 15) << 4;
  const int m = lane & 15;
  const int kg = lane >> 4;

  const float* y1r = Y1r + (size_t)slice * SLICE_ELEMS;
  const float* y1i = Y1i + (size_t)slice * SLICE_ELEMS;

  v8f cr = {};
  v8f ci = {};
  for (int k = 0; k < NN; k += 4) {
    const int ko = k + 2 * kg;
    // A: M[th+m][ko..ko+1]
    v2f ar = *(const v2f*)(Mr + (th + m) * NN + ko);
    v2f ai = *(const v2f*)(Mi + (th + m) * NN + ko);
    v2f ain = -ai;  // f32 WMMA has no A-neg modifier; negate in VALU
    // B: B[kk][n] = Y1[k+kk][tw+n] (strided by a row)
    v2f br, bi;
    br.x = y1r[ko * NN + tw + m];
    br.y = y1r[(ko + 1) * NN + tw + m];
    bi.x = y1i[ko * NN + tw + m];
    bi.y = y1i[(ko + 1) * NN + tw + m];
    cr = wmma_f32(ar, br, cr);
    cr = wmma_f32(ain, bi, cr);
    ci = wmma_f32(ar, bi, ci);
    ci = wmma_f32(ai, br, ci);
  }

  float* y2r = Y2r + (size_t)slice * SLICE_ELEMS;
  float* y2i = Y2i + (size_t)slice * SLICE_ELEMS;
#pragma unroll
  for (int r = 0; r < 8; ++r) {
    const int row = th + r + 8 * kg;
    y2r[row * NN + tw + m] = cr[r];
    y2i[row * NN + tw + m] = ci[r];
  }
}

// ---------------------------------------------------------------------------
// Kernel 4: fused magnitude/phase feature + 1x1 conv as WMMA GEMM.
// out(32 x pixels) = W(32x64) * Feat(64 x pixels), Feat staged in LDS.
// Block = 64 threads (2 waves): wave0 -> out rows 0..15, wave1 -> 16..31.
// ---------------------------------------------------------------------------
__global__ __launch_bounds__(64) void feat_conv_kernel(
    const float* __restrict__ Y2r, const float* __restrict__ Y2i,
    const float* __restrict__ Wmat, float* __restrict__ out) {
  __shared__ float feat[64][16];  // rows 0..31 mag, 32..63 angle/pi

  const int tid = threadIdx.x;
  const int lane = tid & 31;
  const int wave = tid >> 5;
  const int tileLin = blockIdx.x;       // NB * 4096 tiles of 16 pixels
  const int b = tileLin >> 12;          // 65536/16 = 4096 tiles per image
  const int pix0 = (tileLin & 4095) << 4;

  // Each thread produces one feature row (16 pixels) into LDS.
  const int ch = tid & 31;
  const float* yr = Y2r + (size_t)(b * NC + ch) * SLICE_ELEMS + pix0;
  const float* yi = Y2i + (size_t)(b * NC + ch) * SLICE_ELEMS + pix0;
  if (tid < 32) {
#pragma unroll
    for (int px = 0; px < 16; ++px) {
      float re = yr[px], im = yi[px];
      feat[tid][px] = sqrtf(re * re + im * im);
    }
  } else {
#pragma unroll
    for (int px = 0; px < 16; ++px) {
      float re = yr[px], im = yi[px];
      feat[tid][px] = atan2f(im, re) * (1.0f / PI_F);
    }
  }
  __syncthreads();

  const int m0 = wave << 4;  // output-row block
  const int m = lane & 15;
  const int kg = lane >> 4;
  v8f c = {};
  for (int k = 0; k < 64; k += 4) {
    const int ko = k + 2 * kg;
    // A: W[m0+m][ko..ko+1] (row-major 32x64, contiguous v2f)
    v2f a = *(const v2f*)(Wmat + (m0 + m) * 64 + ko);
    // B: Feat[ko..ko+1][pixel m] from LDS
    v2f bf;
    bf.x = feat[ko][m];
    bf.y = feat[ko + 1][m];
    c = wmma_f32(a, bf, c);
  }

  float* ob = out + (size_t)b * NC * SLICE_ELEMS;
#pragma unroll
  for (int r = 0; r < 8; ++r) {
    ob[(size_t)(m0 + r + 8 * kg) * SLICE_ELEMS + pix0 + m] = c[r];
  }
}

// ---------------------------------------------------------------------------
extern "C" void kernel_launch(void* const* d_in, const int* in_sizes, int n_in,
                              void* d_out, int out_size, void* d_ws, size_t ws_size,
                              hipStream_t stream) {
  const float* x = (const float*)d_in[0];      // (4,32,256,256) f32
  const float* alpha = (const float*)d_in[1];  // scalar
  const float* w = (const float*)d_in[2];      // (32,64) f32
  float* out = (float*)d_out;                  // (4,32,256,256) f32

  // Workspace layout (all f32 planes):
  //   Mr, Mi         : 65536 each
  //   Y1r, Y1i       : 128*65536 each
  //   Y2r, Y2i       : 128*65536 each
  char* ws = (char*)d_ws;
  size_t off = 0;
  float* Mr = (float*)(ws + off);  off += (size_t)SLICE_ELEMS * sizeof(float);
  float* Mi = (float*)(ws + off);  off += (size_t)SLICE_ELEMS * sizeof(float);
  float* Y1r = (float*)(ws + off); off += (size_t)NSLICE * SLICE_ELEMS * sizeof(float);
  float* Y1i = (float*)(ws + off); off += (size_t)NSLICE * SLICE_ELEMS * sizeof(float);
  float* Y2r = (float*)(ws + off); off += (size_t)NSLICE * SLICE_ELEMS * sizeof(float);
  float* Y2i = (float*)(ws + off); off += (size_t)NSLICE * SLICE_ELEMS * sizeof(float);

  // 1) Build FRFT operator M (256x256 complex).
  build_M_kernel<<<SLICE_ELEMS / 256, 256, 0, stream>>>(alpha, Mr, Mi);

  // 2) Width pass: Y1 = X * M^T  (256 tiles/slice, 4 waves/block).
  dim3 grid_gemm(64, NSLICE);
  frft_pass1<<<grid_gemm, 128, 0, stream>>>(x, Mr, Mi, Y1r, Y1i);

  // 3) Height pass: Y2 = M * Y1 (complex GEMM).
  frft_pass2<<<grid_gemm, 128, 0, stream>>>(Mr, Mi, Y1r, Y1i, Y2r, Y2i);

  // 4) Magnitude/phase features + 1x1 conv (WMMA GEMM over channels).
  feat_conv_kernel<<<NB * 4096, 64, 0, stream>>>(Y2r, Y2i, w, out);
}